// dgl_GIN_10282151707717
// MI455X (gfx1250) — compile-verified
//
#include <hip/hip_runtime.h>
#include <hip/hip_bf16.h>
#include <stdint.h>

// ---------------------------------------------------------------------------
// GIN forward on gfx1250 (MI455X). fp32 end-to-end using V_WMMA_F32_16X16X4_F32
// (exact fp32 matrix math; workload is L2/atomic bound). A-tiles are staged
// with GLOBAL_LOAD_ASYNC_TO_LDS_B128 + s_wait_asynccnt double buffering when
// no elementwise preprocess is fused; LDS layouts are bank-conflict-free and
// give single ds_load_b64 per WMMA fragment.
// ---------------------------------------------------------------------------

typedef __attribute__((ext_vector_type(2))) float v2f;
typedef __attribute__((ext_vector_type(8))) float v8f;

#define DIM 256          // IN_DIM == HID_DIM
#define ODIM 64
#define TM 64
#define TN 64
#define TK 32
#define ASTRIDE 36       // A tile row stride (floats): 16B aligned, conflict-free
#define BSTRIDE 160      // B tile pair-row stride (floats): == 32 banks -> conflict-free
#define NPART 128        // partial-reduction blocks for BN stats
#define BN_EPS 1e-5f

// ----------------------- CDNA5 async global->LDS ---------------------------
__device__ __forceinline__ void async_load_b128(uint32_t lds_off, const float* gaddr) {
    asm volatile("global_load_async_to_lds_b128 %0, %1, off"
                 :: "v"(lds_off), "v"(gaddr) : "memory");
}
__device__ __forceinline__ void wait_asynccnt0() {
    asm volatile("s_wait_asynccnt 0x0" ::: "memory");
}

// ------------------------------ copy: z = x --------------------------------
__global__ __launch_bounds__(256)
void gin_copy_f4(const float4* __restrict__ x, float4* __restrict__ z, long long n4) {
    long long i = (long long)blockIdx.x * 256 + threadIdx.x;
    if (i < n4) z[i] = x[i];
}

// -------------------- edge scatter: z[dst] += x[src] -----------------------
__global__ __launch_bounds__(256)
void gin_edge_scatter(const float* __restrict__ x, const int* __restrict__ src,
                      const int* __restrict__ dst, float* __restrict__ z, int nE) {
    long long tid = (long long)blockIdx.x * 256 + threadIdx.x;
    int e = (int)(tid >> 6);
    if (e >= nE) return;
    int c4 = (int)(tid & 63);
    int s = src[e], d = dst[e];
    float4 v = ((const float4*)(x + (size_t)s * DIM))[c4];
    float* zp = z + (size_t)d * DIM + c4 * 4;
    atomicAdd(zp + 0, v.x);
    atomicAdd(zp + 1, v.y);
    atomicAdd(zp + 2, v.z);
    atomicAdd(zp + 3, v.w);
}

// --------------------------- WMMA fp32 GEMM --------------------------------
// C[M,N] (+)= pre(A)[M,K] * B[K,N] (+ bias)
// pre(a) = aScale ? max(0, a*aScale[k] + aShift[k]) : a   (fused BN+ReLU)
// Block: 128 threads (4 waves). Tile 64x64, K-chunk 32, double-buffered.
// A staged via global_load_async_to_lds_b128 when no preprocess is needed.
// B staged k-pair-interleaved: Bs[(k>>1)*BSTRIDE + 2n + (k&1)] so each WMMA
// B fragment is one aligned ds_load_b64.
__global__ __launch_bounds__(128)
void gin_gemm_wmma(const float* __restrict__ A, const float* __restrict__ B,
                   float* __restrict__ C, int M, int K, int N,
                   const float* __restrict__ aScale, const float* __restrict__ aShift,
                   const float* __restrict__ bias, int accumulate) {
    __shared__ float As[2][TM * ASTRIDE];      // 2 x 2304 floats
    __shared__ float Bs[2][(TK / 2) * BSTRIDE]; // 2 x 2560 floats

    const int tid  = threadIdx.x;
    const int wave = tid >> 5;
    const int lane = tid & 31;
    const int r    = lane & 15;   // M index (A), N index (B/C) within 16
    const int hi   = lane >> 4;   // upper half-wave (K offset +2)
    const int m0   = blockIdx.y * TM;
    const int n0   = blockIdx.x * TN;
    const bool pre = (aScale != nullptr);

    const uint32_t asBase0 = (uint32_t)(uintptr_t)&As[0][0];
    const uint32_t asBase1 = (uint32_t)(uintptr_t)&As[1][0];

    v8f acc[4];
    acc[0] = (v8f){}; acc[1] = (v8f){}; acc[2] = (v8f){}; acc[3] = (v8f){};

    // per-thread staging coordinates (4 x float4 chunks each for A and B)
    // A: 512 chunks = 64 rows x 8;  B: 512 chunks = 32 rows x 16
    const int aRow[4] = { (tid + 0) >> 3, (tid + 128) >> 3, (tid + 256) >> 3, (tid + 384) >> 3 };
    const int aKq     = tid & 7;
    const int bRow[4] = { (tid + 0) >> 4, (tid + 128) >> 4, (tid + 256) >> 4, (tid + 384) >> 4 };
    const int bNq     = tid & 15;

    auto stageA = [&](int buf, int k0) {
        if (pre) {   // synchronous load + BN/ReLU transform + ds_store_b128
            #pragma unroll
            for (int i = 0; i < 4; ++i) {
                int gm = m0 + aRow[i];
                float4 v = make_float4(0.f, 0.f, 0.f, 0.f);
                if (gm < M) v = *(const float4*)(A + (size_t)gm * K + k0 + aKq * 4);
                int kb = k0 + aKq * 4;
                v.x = fmaxf(0.f, v.x * aScale[kb + 0] + aShift[kb + 0]);
                v.y = fmaxf(0.f, v.y * aScale[kb + 1] + aShift[kb + 1]);
                v.z = fmaxf(0.f, v.z * aScale[kb + 2] + aShift[kb + 2]);
                v.w = fmaxf(0.f, v.w * aScale[kb + 3] + aShift[kb + 3]);
                *(float4*)&As[buf][aRow[i] * ASTRIDE + aKq * 4] = v;
            }
        } else {     // async DMA straight into LDS (16B chunks, padded layout)
            uint32_t base = buf ? asBase1 : asBase0;
            #pragma unroll
            for (int i = 0; i < 4; ++i) {
                int gm = m0 + aRow[i];
                if (gm < M) {   // skipped rows leave stale LDS; their C rows are never stored
                    uint32_t lds = base + (uint32_t)(aRow[i] * ASTRIDE + aKq * 4) * 4u;
                    async_load_b128(lds, A + (size_t)gm * K + k0 + aKq * 4);
                }
            }
        }
    };
    auto loadB = [&](float4* breg, int k0) {
        #pragma unroll
        for (int i = 0; i < 4; ++i)
            breg[i] = *(const float4*)(B + (size_t)(k0 + bRow[i]) * N + n0 + bNq * 4);
    };
    auto storeB = [&](const float4* breg, int buf) {
        #pragma unroll
        for (int i = 0; i < 4; ++i) {
            int k = bRow[i];
            int base = (k >> 1) * BSTRIDE + (bNq * 4) * 2 + (k & 1);
            Bs[buf][base + 0] = breg[i].x;
            Bs[buf][base + 2] = breg[i].y;
            Bs[buf][base + 4] = breg[i].z;
            Bs[buf][base + 6] = breg[i].w;
        }
    };

    // ---- prologue: stage chunk 0 into buffer 0 ---------------------------
    float4 breg[4];
    stageA(0, 0);
    loadB(breg, 0);
    storeB(breg, 0);
    wait_asynccnt0();
    __syncthreads();

    const int nch = K / TK;
    for (int c = 0; c < nch; ++c) {
        const int buf  = c & 1;
        const int nxt  = buf ^ 1;
        const bool more = (c + 1 < nch);

        if (more) {
            stageA(nxt, (c + 1) * TK);   // async: overlaps with compute below
            loadB(breg, (c + 1) * TK);   // VMEM latency hides under WMMAs
        }

        // ---- 8 k-steps x 4 N-tiles = 32 WMMAs per wave per chunk ---------
        const float* asrow = &As[buf][(wave * 16 + r) * ASTRIDE];
        #pragma unroll
        for (int kk = 0; kk < TK; kk += 4) {
            v2f a = *(const v2f*)(asrow + kk + 2 * hi);                 // ds_load_b64
            const float* bbase = &Bs[buf][((kk >> 1) + hi) * BSTRIDE + 2 * r];
            v2f b0 = *(const v2f*)(bbase + 0);                          // ds_load_b64
            v2f b1 = *(const v2f*)(bbase + 32);
            v2f b2 = *(const v2f*)(bbase + 64);
            v2f b3 = *(const v2f*)(bbase + 96);
            acc[0] = __builtin_amdgcn_wmma_f32_16x16x4_f32(false, a, false, b0, (short)0, acc[0], false, false);
            acc[1] = __builtin_amdgcn_wmma_f32_16x16x4_f32(false, a, false, b1, (short)0, acc[1], false, false);
            acc[2] = __builtin_amdgcn_wmma_f32_16x16x4_f32(false, a, false, b2, (short)0, acc[2], false, false);
            acc[3] = __builtin_amdgcn_wmma_f32_16x16x4_f32(false, a, false, b3, (short)0, acc[3], false, false);
        }

        if (more) storeB(breg, nxt);     // after compute: B[c+1] -> next buffer
        wait_asynccnt0();                // A[c+1] landed in LDS
        __syncthreads();                 // all waves done with buf; stores visible
    }

    // ---- epilogue: optional bias, optional accumulate into C -------------
    #pragma unroll
    for (int t = 0; t < 4; ++t) {
        int col = n0 + t * 16 + r;
        #pragma unroll
        for (int e = 0; e < 8; ++e) {
            int gm = m0 + wave * 16 + e + 8 * hi;   // C/D VGPR e -> M = e + 8*hi
            if (gm < M) {
                float val = acc[t][e];
                if (bias) val += bias[col];
                float* p = C + (size_t)gm * N + col;
                if (accumulate) val += *p;
                *p = val;
            }
        }
    }
}

// ------------------- BN stats: deterministic 2-pass ------------------------
__global__ __launch_bounds__(256)
void gin_bn_partial(const float* __restrict__ X, float* __restrict__ P, int M) {
    int c = threadIdx.x;                  // column 0..255
    float s = 0.f, s2 = 0.f;
    for (int row = blockIdx.x; row < M; row += gridDim.x) {
        float v = X[(size_t)row * DIM + c];
        s += v; s2 += v * v;
    }
    P[(size_t)blockIdx.x * 512 + c]       = s;
    P[(size_t)blockIdx.x * 512 + 256 + c] = s2;
}

__global__ __launch_bounds__(256)
void gin_bn_finalize(const float* __restrict__ P, int M,
                     const float* __restrict__ gamma, const float* __restrict__ beta,
                     float* __restrict__ ss) {   // ss[0..255]=scale, ss[256..511]=shift
    int c = threadIdx.x;
    float s = 0.f, s2 = 0.f;
    for (int b = 0; b < NPART; ++b) {
        s  += P[(size_t)b * 512 + c];
        s2 += P[(size_t)b * 512 + 256 + c];
    }
    float invM  = 1.f / (float)M;
    float mean  = s * invM;
    float var   = s2 * invM - mean * mean;
    float rstd  = rsqrtf(var + BN_EPS);
    float scale = gamma[c] * rstd;
    ss[c]       = scale;
    ss[256 + c] = beta[c] - mean * scale;
}

// ----------------- BN+ReLU apply (materialize hidden rep) ------------------
__global__ __launch_bounds__(256)
void gin_bn_relu_apply(const float4* __restrict__ X, const float* __restrict__ ss,
                       float4* __restrict__ Y, long long n4) {
    long long i = (long long)blockIdx.x * 256 + threadIdx.x;
    if (i >= n4) return;
    int cb = (int)((i * 4) & (DIM - 1));
    float4 v = X[i];
    v.x = fmaxf(0.f, v.x * ss[cb + 0] + ss[256 + cb + 0]);
    v.y = fmaxf(0.f, v.y * ss[cb + 1] + ss[256 + cb + 1]);
    v.z = fmaxf(0.f, v.z * ss[cb + 2] + ss[256 + cb + 2]);
    v.w = fmaxf(0.f, v.w * ss[cb + 3] + ss[256 + cb + 3]);
    Y[i] = v;
}

// ---------------------------------------------------------------------------
static inline void launch_gemm(const float* A, const float* B, float* C,
                               int M, int K, int N,
                               const float* aScale, const float* aShift,
                               const float* bias, int accumulate,
                               hipStream_t stream) {
    dim3 grid(N / TN, (M + TM - 1) / TM);
    gin_gemm_wmma<<<grid, 128, 0, stream>>>(A, B, C, M, K, N,
                                            aScale, aShift, bias, accumulate);
}

extern "C" void kernel_launch(void* const* d_in, const int* in_sizes, int n_in,
                              void* d_out, int out_size, void* d_ws, size_t ws_size,
                              hipStream_t stream) {
    const float* h   = (const float*)d_in[0];
    const int*   src = (const int*)d_in[1];
    const int*   dst = (const int*)d_in[2];
    const int M  = in_sizes[0] / DIM;     // 50000
    const int nE = in_sizes[1];           // 800000

    // gin layer params: base 3, six per layer (w1,w2,g1,b1,g2,b2); lp from 15
    const size_t mn = (size_t)M * DIM;
    float* ws = (float*)d_ws;
    float* X1 = ws;                       // hidden rep layer 1
    float* X2 = X1 + mn;                  // hidden rep layer 2
    float* Z  = X2 + mn;                  // z = x + agg ; later reused for T2
    float* T  = Z  + mn;                  // T1 = z @ w1
    float* P  = T  + mn;                  // NPART x 512 partials
    float* SS = P  + (size_t)NPART * 512; // scale/shift (512)

    const long long n4   = (long long)mn / 4;
    const int grid_ew    = (int)(((long long)nE * 64 + 255) / 256);
    const int grid_elem  = (int)((n4 + 255) / 256);

    const float* x = h;
    float* hidden[2] = {X1, X2};

    for (int l = 0; l < 2; ++l) {
        const float* w1 = (const float*)d_in[3 + l * 6 + 0];
        const float* w2 = (const float*)d_in[3 + l * 6 + 1];
        const float* g1 = (const float*)d_in[3 + l * 6 + 2];
        const float* b1 = (const float*)d_in[3 + l * 6 + 3];
        const float* g2 = (const float*)d_in[3 + l * 6 + 4];
        const float* b2 = (const float*)d_in[3 + l * 6 + 5];

        // z = x + segment_sum(x[src], dst)
        gin_copy_f4<<<grid_elem, 256, 0, stream>>>((const float4*)x, (float4*)Z, n4);
        gin_edge_scatter<<<grid_ew, 256, 0, stream>>>(x, src, dst, Z, nE);

        // T = z @ w1   (async A staging)
        launch_gemm(Z, w1, T, M, DIM, DIM, nullptr, nullptr, nullptr, 0, stream);

        // BN1 stats on T -> SS
        gin_bn_partial<<<NPART, 256, 0, stream>>>(T, P, M);
        gin_bn_finalize<<<1, 256, 0, stream>>>(P, M, g1, b1, SS);

        // Z(reused) = relu(bn1(T)) @ w2   (BN+ReLU fused into sync A staging)
        launch_gemm(T, w2, Z, M, DIM, DIM, SS, SS + 256, nullptr, 0, stream);

        // BN2 stats on Z -> SS ; hidden[l] = relu(bn2(Z))
        gin_bn_partial<<<NPART, 256, 0, stream>>>(Z, P, M);
        gin_bn_finalize<<<1, 256, 0, stream>>>(P, M, g2, b2, SS);
        gin_bn_relu_apply<<<grid_elem, 256, 0, stream>>>((const float4*)Z, SS,
                                                         (float4*)hidden[l], n4);
        x = hidden[l];
    }

    // score = sum_i hidden_rep[i] @ lp[i].w + lp[i].b   (async A staging)
    float* out = (float*)d_out;
    const float* lp0w = (const float*)d_in[15];
    const float* lp0b = (const float*)d_in[16];
    const float* lp1w = (const float*)d_in[17];
    const float* lp1b = (const float*)d_in[18];
    const float* lp2w = (const float*)d_in[19];
    const float* lp2b = (const float*)d_in[20];

    launch_gemm(h,  lp0w, out, M, DIM, ODIM, nullptr, nullptr, lp0b, 0, stream);
    launch_gemm(X1, lp1w, out, M, DIM, ODIM, nullptr, nullptr, lp1b, 1, stream);
    launch_gemm(X2, lp2w, out, M, DIM, ODIM, nullptr, nullptr, lp2b, 1, stream);
}